// GraphConvLayer_89567247990813
// MI455X (gfx1250) — compile-verified
//
#include <hip/hip_runtime.h>
#include <hip/hip_bf16.h>

typedef __attribute__((ext_vector_type(2))) float v2f;
typedef __attribute__((ext_vector_type(8))) float v8f;

#define DDIM 128

// ---------------------------------------------------------------------------
// Kernel 1: zero the aggregation buffer (float4 grid-stride memset)
// ---------------------------------------------------------------------------
__global__ void gc_zero_kernel(float4* __restrict__ agg4, int nvec) {
    int i = blockIdx.x * blockDim.x + threadIdx.x;
    int stride = gridDim.x * blockDim.x;
    for (; i < nvec; i += stride) {
        agg4[i] = make_float4(0.f, 0.f, 0.f, 0.f);
    }
}

// ---------------------------------------------------------------------------
// Kernel 2: scatter-add  agg[row] += x[col]
// One wave (32 lanes) per edge; each lane moves a float4 chunk of the 128-dim
// row: float4 gather from x (L2-resident, 51MB) + 4 fp32 global atomics.
// ---------------------------------------------------------------------------
__global__ void gc_scatter_kernel(const float* __restrict__ x,
                                  const long long* __restrict__ eidx,
                                  float* __restrict__ agg,
                                  int E) {
    int gid = blockIdx.x * blockDim.x + threadIdx.x;
    int e = gid >> 5;          // edge handled by this wave
    int lane = gid & 31;       // chunk within the 128-float row
    if (e >= E) return;
    // edge_index is [2, E] flat: row = eidx[e], col = eidx[E + e]
    long long row = eidx[e];
    long long col = eidx[(long long)E + e];
    int c = lane << 2;         // 4 floats per lane
    const float4 v = *(const float4*)(x + col * DDIM + c);
    float* dst = agg + row * DDIM + c;
    unsafeAtomicAdd(dst + 0, v.x);   // -> global_atomic_add_f32
    unsafeAtomicAdd(dst + 1, v.y);
    unsafeAtomicAdd(dst + 2, v.z);
    unsafeAtomicAdd(dst + 3, v.w);
}

// ---------------------------------------------------------------------------
// Kernel 3: h = (x + agg) @ W^T + b ; out = silu(h)
// fp32 WMMA (V_WMMA_F32_16X16X4_F32). 256 threads = 8 waves; each wave owns a
// 16-row M tile and all 128 output columns (8 accumulators of 16x16).
// W (128x128 fp32, 64KB) staged in LDS with XOR swizzle to avoid 16-way bank
// conflicts. B fragments for all 8 N-tiles are loaded into registers FIRST,
// then the 8 WMMAs issue back-to-back: one s_wait_dscnt per k-step instead of
// one per WMMA, overlapping DS latency with the matrix pipe.
// ---------------------------------------------------------------------------
__global__ void gc_gemm_silu_kernel(const float* __restrict__ x,
                                    const float* __restrict__ agg,
                                    const float* __restrict__ W,
                                    const float* __restrict__ bias,
                                    float* __restrict__ out,
                                    int N) {
    extern __shared__ float lds[];   // 128 * 128 floats = 64 KB (swizzled)

    const int t = threadIdx.x;       // 0..255
    // ---- cooperative load of W into LDS (swizzled: phys_k = k ^ ((row&15)<<3))
    #pragma unroll 4
    for (int i = t; i < DDIM * (DDIM / 4); i += 256) {
        int r  = i >> 5;             // W row (output channel), 0..127
        int c4 = (i & 31) << 2;      // column group of 4
        const float4 w4 = *(const float4*)(W + r * DDIM + c4);
        int phys = c4 ^ ((r & 15) << 3);  // bits >=3 -> float4 stays contiguous
        *(float4*)(lds + r * DDIM + phys) = w4;
    }
    __syncthreads();

    const int wave = t >> 5;                 // 0..7
    const int lane = t & 31;
    const int mloc = lane & 15;              // row within 16-row tile (A/B frag)
    const int kgrp = lane >> 4;              // K sub-group (0/1)
    const long long row_base = (long long)blockIdx.x * 128 + wave * 16;

    v8f acc[8] = {};                         // 8 N-tiles of 16x16 fp32

    // clamp row for loads (WMMA requires EXEC all-ones; stores are predicated)
    long long r = row_base + mloc;
    long long rc = (r < (long long)N) ? r : (long long)(N - 1);
    const float* xrow = x   + rc * DDIM;
    const float* grow = agg + rc * DDIM;

    // swizzled LDS base for this lane: row (nt*16 + mloc), col (kk ^ mloc<<3)
    const float* ldsrow = lds + mloc * DDIM; // + nt*0x2000/4 + (kk ^ swz)
    const int swz = mloc << 3;

    #pragma unroll 4
    for (int ks = 0; ks < 32; ++ks) {        // K = 128 in steps of 4
        int kk = (ks << 2) + (kgrp << 1);    // even -> 8B-aligned float2
        // A fragment: (x + agg), contiguous pair, reused across 8 N-tiles
        v2f xa = *(const v2f*)(xrow + kk);
        v2f ga = *(const v2f*)(grow + kk);
        v2f a;
        a.x = xa.x + ga.x;
        a.y = xa.y + ga.y;
        // ---- batch all 8 B-fragment LDS loads (single dscnt wait)
        v2f bf[8];
        const int phys = kk ^ swz;           // same for every nt (n&15 == mloc)
        #pragma unroll
        for (int nt = 0; nt < 8; ++nt) {
            bf[nt] = *(const v2f*)(ldsrow + (nt << 11) + phys);
        }
        // ---- 8 back-to-back WMMAs (disjoint accumulators; no D->A/B hazard)
        #pragma unroll
        for (int nt = 0; nt < 8; ++nt) {
            acc[nt] = __builtin_amdgcn_wmma_f32_16x16x4_f32(
                false, a, false, bf[nt], (short)0, acc[nt], false, false);
        }
    }

    // ---- epilogue: + bias, SiLU, predicated store
    #pragma unroll
    for (int nt = 0; nt < 8; ++nt) {
        #pragma unroll
        for (int rr = 0; rr < 8; ++rr) {
            int m = rr + (kgrp << 3);        // D layout: lanes16-31 -> M+8
            int o = (nt << 4) + mloc;        // output column
            long long gr = row_base + m;
            if (gr < (long long)N) {
                float h = acc[nt][rr] + bias[o];
                float s = h / (1.0f + __expf(-h));
                out[gr * DDIM + o] = s;
            }
        }
    }
}

// ---------------------------------------------------------------------------
extern "C" void kernel_launch(void* const* d_in, const int* in_sizes, int n_in,
                              void* d_out, int out_size, void* d_ws, size_t ws_size,
                              hipStream_t stream) {
    const float*     x    = (const float*)d_in[0];
    const long long* eidx = (const long long*)d_in[1];   // edge_index int64 [2,E]
    // d_in[2] = edge_attr: unused by the reference
    const float*     W    = (const float*)d_in[3];       // [128,128]
    const float*     bias = (const float*)d_in[4];       // [128]
    float*           out  = (float*)d_out;

    const int Ddim = in_sizes[4];            // 128
    const int N    = in_sizes[0] / Ddim;     // 100000
    const int E    = in_sizes[1] / 2;        // 1600000

    float* agg = (float*)d_ws;               // N*128 fp32 = 51.2 MB scratch

    // 1) zero agg
    int nvec = (N * Ddim) >> 2;
    gc_zero_kernel<<<1024, 256, 0, stream>>>((float4*)agg, nvec);

    // 2) scatter-add: one wave per edge
    long long sthreads = (long long)E * 32;
    int sblocks = (int)((sthreads + 255) / 256);
    gc_scatter_kernel<<<sblocks, 256, 0, stream>>>(x, eidx, agg, E);

    // 3) fused (x+agg)@W^T + b, SiLU, via fp32 WMMA; 64KB dynamic LDS for W
    int gblocks = (N + 127) / 128;
    gc_gemm_silu_kernel<<<gblocks, 256, DDIM * DDIM * sizeof(float), stream>>>(
        x, agg, W, bias, out, N);
}